// RAConv_49452253446302
// MI455X (gfx1250) — compile-verified
//
#include <hip/hip_runtime.h>
#include <hip/hip_bf16.h>

typedef float v2f __attribute__((ext_vector_type(2)));
typedef float v8f __attribute__((ext_vector_type(8)));

__device__ __forceinline__ void atomAddF(float* p, float v) {
    __hip_atomic_fetch_add(p, v, __ATOMIC_RELAXED, __HIP_MEMORY_SCOPE_AGENT);
}
__device__ __forceinline__ void atomMaxU(unsigned* p, unsigned v) {
    __hip_atomic_fetch_max(p, v, __ATOMIC_RELAXED, __HIP_MEMORY_SCOPE_AGENT);
}

// Monotone float <-> uint mapping so we can use integer atomic max.
__device__ __forceinline__ unsigned encf(float f) {
    unsigned u = __float_as_uint(f);
    return (u & 0x80000000u) ? ~u : (u | 0x80000000u);
}
__device__ __forceinline__ float decf(unsigned e) {
    return (e & 0x80000000u) ? __uint_as_float(e ^ 0x80000000u)
                             : __uint_as_float(~e);
}

__global__ void k_zero(float* __restrict__ p, size_t n) {
    size_t i = (size_t)blockIdx.x * blockDim.x + threadIdx.x;
    size_t st = (size_t)gridDim.x * blockDim.x;
    for (; i < n; i += st) p[i] = 0.0f;
}

// Pass A: one wave per edge. logits[e] = dot(x[src], x[dst]) * D^-0.5 ; seg max.
__global__ void k_edge_logits(const float* __restrict__ x, const int* __restrict__ src,
                              const int* __restrict__ dst, float* __restrict__ logits,
                              unsigned* __restrict__ encmax, int E) {
    int tid  = blockIdx.x * blockDim.x + threadIdx.x;
    int wid  = tid >> 5;
    int lane = tid & 31;
    if (wid >= E) return;
    int s = src[wid], d = dst[wid];
    float2 a = *(const float2*)(x + (size_t)s * 64 + lane * 2);
    float2 b = *(const float2*)(x + (size_t)d * 64 + lane * 2);
    float p = a.x * b.x + a.y * b.y;
    #pragma unroll
    for (int off = 16; off; off >>= 1) p += __shfl_xor(p, off, 32);
    if (lane == 0) {
        float lg = p * 0.125f;  // 64^-0.5
        logits[wid] = lg;
        atomMaxU(encmax + d, encf(lg));
    }
}

// Pass B: ex = exp(logit - max[dst]); denom += ex; cnt += 1; sum_x[dst] += x_src.
__global__ void k_edge_exp(const float* __restrict__ x, const int* __restrict__ src,
                           const int* __restrict__ dst, float* __restrict__ exbuf,
                           const unsigned* __restrict__ encmax, float* __restrict__ denom,
                           float* __restrict__ cnt, float* __restrict__ sum, int E) {
    int tid  = blockIdx.x * blockDim.x + threadIdx.x;
    int wid  = tid >> 5;
    int lane = tid & 31;
    if (wid >= E) return;
    int s = src[wid], d = dst[wid];
    float mx = decf(encmax[d]);
    float ex = __expf(exbuf[wid] - mx);
    if (lane == 0) {
        exbuf[wid] = ex;
        atomAddF(denom + d, ex);
        atomAddF(cnt + d, 1.0f);
    }
    float2 v = *(const float2*)(x + (size_t)s * 64 + lane * 2);
    atomAddF(&sum[(size_t)d * 64 + lane * 2 + 0], v.x);
    atomAddF(&sum[(size_t)d * 64 + lane * 2 + 1], v.y);
}

// mean = sum / max(cnt,1)
__global__ void k_mean(const float* __restrict__ sum, const float* __restrict__ cnt,
                       float* __restrict__ mean, size_t n64) {
    size_t i = (size_t)blockIdx.x * blockDim.x + threadIdx.x;
    if (i >= n64) return;
    mean[i] = sum[i] / fmaxf(cnt[i >> 6], 1.0f);
}

// Pass C: msg[dst] += attn * x_src ; var[dst] += (x_src - mean[dst])^2
__global__ void k_edge_scatter(const float* __restrict__ x, const int* __restrict__ src,
                               const int* __restrict__ dst, const float* __restrict__ exbuf,
                               const float* __restrict__ denom, const float* __restrict__ mean,
                               float* __restrict__ msg, float* __restrict__ var, int E) {
    int tid  = blockIdx.x * blockDim.x + threadIdx.x;
    int wid  = tid >> 5;
    int lane = tid & 31;
    if (wid >= E) return;
    int s = src[wid], d = dst[wid];
    float attn = exbuf[wid] / denom[d];
    size_t o = (size_t)d * 64 + lane * 2;
    float2 v = *(const float2*)(x + (size_t)s * 64 + lane * 2);
    float2 m = *(const float2*)(mean + o);
    atomAddF(&msg[o + 0], attn * v.x);
    atomAddF(&msg[o + 1], attn * v.y);
    float dx = v.x - m.x, dy = v.y - m.y;
    atomAddF(&var[o + 0], dx * dx);
    atomAddF(&var[o + 1], dy * dy);
}

// var /= max(cnt,1)
__global__ void k_varnorm(float* __restrict__ var, const float* __restrict__ cnt, size_t n64) {
    size_t i = (size_t)blockIdx.x * blockDim.x + threadIdx.x;
    if (i >= n64) return;
    var[i] = var[i] / fmaxf(cnt[i >> 6], 1.0f);
}

// out = x@Ws^T + msg@Wn^T + var@Wv^T + (bs+bn+bv), fp32 WMMA 16x16x4.
// Weights staged once per block into LDS, pre-swizzled into B-fragment order:
//   lw[((m*16 + kb)*4 + t)*64 + lane*2 + q] = W_m[(t*16 + (lane&15))*64 + kb*4 + (lane>>4)*2 + q]
// so compute-loop B loads are conflict-free lane-contiguous ds_load_b64.
__global__ void k_out_gemm(const float* __restrict__ x, const float* __restrict__ msg,
                           const float* __restrict__ var,
                           const float* __restrict__ Ws, const float* __restrict__ bs,
                           const float* __restrict__ Wn, const float* __restrict__ bn,
                           const float* __restrict__ Wv, const float* __restrict__ bv,
                           float* __restrict__ out, int N) {
    __shared__ float lw[3 * 16 * 4 * 64];   // 48 KB: [m][kb][t][64]
    const float* Wsrc[3] = {Ws, Wn, Wv};

    // Cooperative staging of B fragments (before any wave can exit -> barrier safe).
    for (int idx = threadIdx.x; idx < 3 * 16 * 4 * 32; idx += blockDim.x) {
        int ln   = idx & 31;
        int frag = idx >> 5;            // (m*16 + kb)*4 + t
        int t    = frag & 3;
        int kb   = (frag >> 2) & 15;
        int m    = frag >> 6;
        int j    = t * 16 + (ln & 15);            // output column
        int k    = kb * 4 + (ln >> 4) * 2;        // K pair
        v2f w2 = *(const v2f*)(Wsrc[m] + (size_t)j * 64 + k);
        *(v2f*)(lw + (size_t)frag * 64 + ln * 2) = w2;
    }
    __syncthreads();

    int lane = threadIdx.x & 31;
    int wid  = blockIdx.x * (blockDim.x >> 5) + (threadIdx.x >> 5);
    int r0   = wid * 16;
    if (r0 >= N) return;
    int half = lane >> 4;   // 0: K pair {0,1}, 1: K pair {2,3}
    int sub  = lane & 15;

    if (r0 + 16 <= N) {     // wave-uniform: EXEC all-1s inside (WMMA requirement)
        const float* A[3] = {x, msg, var};
        v8f acc[4] = {};
        int arow = r0 + sub;
        int kofs = half * 2;
        for (int kb = 0; kb < 16; ++kb) {
            v2f a[3], b[12];
            #pragma unroll
            for (int m = 0; m < 3; ++m)
                a[m] = *(const v2f*)(A[m] + (size_t)arow * 64 + kb * 4 + kofs);
            #pragma unroll
            for (int m = 0; m < 3; ++m) {
                #pragma unroll
                for (int t = 0; t < 4; ++t)
                    b[m * 4 + t] = *(const v2f*)(lw + (size_t)(((m * 16 + kb) * 4 + t)) * 64 + lane * 2);
            }
            // 12 back-to-back WMMAs after one batched wait
            #pragma unroll
            for (int m = 0; m < 3; ++m) {
                #pragma unroll
                for (int t = 0; t < 4; ++t)
                    acc[t] = __builtin_amdgcn_wmma_f32_16x16x4_f32(
                        false, a[m], false, b[m * 4 + t], (short)0, acc[t], false, false);
            }
        }
        #pragma unroll
        for (int t = 0; t < 4; ++t) {
            int c = t * 16 + sub;
            float bias = bs[c] + bn[c] + bv[c];
            #pragma unroll
            for (int v = 0; v < 8; ++v) {
                int row = r0 + v + half * 8;   // ISA C/D layout: VGPR v -> M=v (+8 for hi lanes)
                out[(size_t)row * 64 + c] = acc[t][v] + bias;
            }
        }
    } else {
        // scalar tail (unused for N=100000, kept for generality)
        for (int r = r0 + lane; r < N; r += 32) {
            for (int c = 0; c < 64; ++c) {
                float acc = bs[c] + bn[c] + bv[c];
                for (int k = 0; k < 64; ++k) {
                    acc += x[(size_t)r * 64 + k]   * Ws[(size_t)c * 64 + k];
                    acc += msg[(size_t)r * 64 + k] * Wn[(size_t)c * 64 + k];
                    acc += var[(size_t)r * 64 + k] * Wv[(size_t)c * 64 + k];
                }
                out[(size_t)r * 64 + c] = acc;
            }
        }
    }
}

extern "C" void kernel_launch(void* const* d_in, const int* in_sizes, int n_in,
                              void* d_out, int out_size, void* d_ws, size_t ws_size,
                              hipStream_t stream) {
    const float* x  = (const float*)d_in[0];
    const int*   ei = (const int*)d_in[1];
    const float* Ws = (const float*)d_in[2];
    const float* bs = (const float*)d_in[3];
    const float* Wn = (const float*)d_in[4];
    const float* bn = (const float*)d_in[5];
    const float* Wv = (const float*)d_in[6];
    const float* bv = (const float*)d_in[7];
    float*       out = (float*)d_out;

    const int N = in_sizes[0] / 64;
    const int E = in_sizes[1] / 2;
    const int* src = ei;
    const int* dst = ei + E;

    // workspace layout (floats)
    float* w        = (float*)d_ws;
    unsigned* encmx = (unsigned*)w;                 // N
    float* denom    = w + (size_t)N;                // N
    float* cnt      = w + 2 * (size_t)N;            // N
    float* exbuf    = w + 3 * (size_t)N;            // E (logits then ex)
    float* sum      = exbuf + (size_t)E;            // N*64
    float* mean     = sum + (size_t)N * 64;         // N*64
    float* msg      = mean + (size_t)N * 64;        // N*64
    float* var      = msg + (size_t)N * 64;         // N*64
    size_t zfloats  = 3 * (size_t)N + (size_t)E + 4 * (size_t)N * 64;

    k_zero<<<2048, 256, 0, stream>>>(w, zfloats);

    int edgeBlocks = (E + 7) / 8;   // 8 waves (256 threads) per block, 1 wave per edge
    k_edge_logits<<<edgeBlocks, 256, 0, stream>>>(x, src, dst, exbuf, encmx, E);
    k_edge_exp<<<edgeBlocks, 256, 0, stream>>>(x, src, dst, exbuf, encmx, denom, cnt, sum, E);

    size_t n64 = (size_t)N * 64;
    int elemBlocks = (int)((n64 + 255) / 256);
    k_mean<<<elemBlocks, 256, 0, stream>>>(sum, cnt, mean, n64);

    k_edge_scatter<<<edgeBlocks, 256, 0, stream>>>(x, src, dst, exbuf, denom, mean, msg, var, E);
    k_varnorm<<<elemBlocks, 256, 0, stream>>>(var, cnt, n64);

    int waves = (N + 15) / 16;
    int gemmBlocks = (waves + 3) / 4;   // 4 waves (128 threads) per block
    k_out_gemm<<<gemmBlocks, 128, 0, stream>>>(x, msg, var, Ws, bs, Wn, bn, Wv, bv, out, N);
}